// FourierCrossAttention_74972949119502
// MI455X (gfx1250) — compile-verified
//
#include <hip/hip_runtime.h>
#include <math.h>

typedef float v2f __attribute__((ext_vector_type(2)));
typedef float v8f __attribute__((ext_vector_type(8)));

// ---------------- problem constants ----------------
#define NB 32
#define NH 8
#define NE 64
#define NM 64
#define NL 1024
#define NO 64

// workspace layout (float offsets)
#define OFF_FC     0                       // [1024][64] cos
#define OFF_FS     65536                   // [1024][64] -sin
#define OFF_G      131072                  // [128][1024] scaled inverse basis
#define OFF_XQT_RE 262144                  // [bh][m][e]
#define PLANE      1048576
#define OFF_XQT_IM (OFF_XQT_RE + 1*PLANE)
#define OFF_XK_RE  (OFF_XQT_RE + 2*PLANE)  // [bh][e][m]
#define OFF_XK_IM  (OFF_XQT_RE + 3*PLANE)
#define OFF_T_RE   (OFF_XQT_RE + 4*PLANE)  // [bh][x][y] tanh(qk)
#define OFF_T_IM   (OFF_XQT_RE + 5*PLANE)
#define OFF_V_RE   (OFF_XQT_RE + 6*PLANE)  // [h][x][b][e]
#define OFF_V_IM   (OFF_XQT_RE + 7*PLANE)
#define OFF_Y      (OFF_XQT_RE + 8*PLANE)  // [b][h][o][128] (re 0..63, im 64..127)

// ---------------- WMMA helpers (f32 16x16x4) ----------------
__device__ __forceinline__ v8f wmma4(v2f a, v2f b, v8f c) {
  return __builtin_amdgcn_wmma_f32_16x16x4_f32(false, a, false, b, (short)0, c,
                                               false, false);
}
__device__ __forceinline__ v2f vneg(v2f a) { v2f r; r.x = -a.x; r.y = -a.y; return r; }

// ---------------- CDNA5 async global->LDS copy (ASYNCcnt path) -------------
// The LDS destination pointer itself is passed into the asm (captured), so
// the compiler knows the asm writes the shared array and keeps the ds_loads
// that consume it alive. Low 32 bits of a generic pointer to LDS == LDS
// byte offset on AMDGPU.
__device__ __forceinline__ void async_load_b128(const float* lds_ptr,
                                                const void* gaddr) {
  asm volatile("global_load_async_to_lds_b128 %0, %1, off"
               :: "v"((unsigned)(unsigned long long)(uintptr_t)lds_ptr),
                  "v"((unsigned long long)(uintptr_t)gaddr)
               : "memory");
}
__device__ __forceinline__ void wait_async0() {
  asm volatile("s_wait_asynccnt 0x0" ::: "memory");
}

// A fragment (16x4 MxK): row = row0 + (lane&15), k = k0 + 2*(lane>>4) + v
__device__ __forceinline__ v2f load_a(const float* __restrict__ base, int ld,
                                      int row0, int k0, int lo, int hi) {
  const float* p = base + (row0 + lo) * ld + k0 + 2 * hi;
  v2f a; a.x = p[0]; a.y = p[1]; return a;
}
// B fragment (4x16 KxN): col = col0 + (lane&15), k = k0 + 2*(lane>>4) + v
__device__ __forceinline__ v2f load_b(const float* __restrict__ base, int ld,
                                      int k0, int col0, int lo, int hi) {
  v2f b;
  b.x = base[(k0 + 2 * hi + 0) * ld + col0 + lo];
  b.y = base[(k0 + 2 * hi + 1) * ld + col0 + lo];
  return b;
}
// B fragment where the source matrix is stored transposed: B[k][n] = baseT[n][k]
__device__ __forceinline__ v2f load_bT(const float* __restrict__ baseT, int ld,
                                       int k0, int col0, int lo, int hi) {
  const float* p = baseT + (col0 + lo) * ld + k0 + 2 * hi;
  v2f b; b.x = p[0]; b.y = p[1]; return b;
}

// complex tanh: tanh(a+ib) = (sinh 2a + i sin 2b)/(cosh 2a + cos 2b)
__device__ __forceinline__ void ctanh(float a, float b, float& tr, float& ti) {
  float a2 = fminf(fmaxf(2.f * a, -40.f), 40.f);
  float ea = __expf(a2), ei = 1.f / ea;
  float sh = 0.5f * (ea - ei), ch = 0.5f * (ea + ei);
  float sb, cb; __sincosf(2.f * b, &sb, &cb);
  float inv = 1.f / (ch + cb);
  tr = sh * inv; ti = sb * inv;
}

// ================= K0: DFT / iDFT bases =================
__global__ void k0_basis(float* __restrict__ ws) {
  const float W = 6.283185307179586f / 1024.f;
  int idx = blockIdx.x * 256 + threadIdx.x;
  if (idx < 65536) {  // forward basis [t][m]
    int t = idx >> 6, m = idx & 63;
    int ph = (m * t) & 1023;
    float s, c; __sincosf(W * (float)ph, &s, &c);
    ws[OFF_FC + idx] = c;
    ws[OFF_FS + idx] = -s;
  }
  if (idx < 131072) {  // inverse basis [r][t], scaled
    int r = idx >> 10, t = idx & 1023;
    int m = r & 63;
    int ph = (m * t) & 1023;
    float s, c; __sincosf(W * (float)ph, &s, &c);
    float sc = ((m == 0) ? 1.f : 2.f) / (1024.f * 262144.f);  // 1/(L*IN*OUT)
    ws[OFF_G + idx] = (r < 64) ? sc * c : -sc * s;
  }
}

// ================= K1: 64-mode DFT of q and k (GEMM) =================
// blockIdx.x = 2*bh + isk ; block computes Xft(E=64 x M=64) complex.
// q/k chunks are staged into LDS with double-buffered async b128 copies.
__global__ void k1_dft(const float* __restrict__ q, const float* __restrict__ kk,
                       float* __restrict__ ws) {
  __shared__ float xs[2 * 2048];  // two 32t x 64e buffers (16 KB)
  int bh = blockIdx.x >> 1, isk = blockIdx.x & 1;
  int b = bh >> 3, h = bh & 7;
  const float* src = isk ? kk : q;
  int base = b * 524288 + h * 64;  // + t*512 + e
  int tid = threadIdx.x;

  int lane = tid & 31, w = tid >> 5;
  int lo = lane & 15, hi = lane >> 4;
  int et = w >> 1;             // e tile 0..3
  int m0 = (w & 1) * 32, m1 = m0 + 16;
  const float* FC = ws + OFF_FC;
  const float* FS = ws + OFF_FS;

  // issue chunk 0 into buffer 0
#pragma unroll
  for (int i = 0; i < 2; ++i) {
    int lin4 = i * 256 + tid;                 // 0..511 float4s
    int tl = lin4 >> 4, e4 = lin4 & 15;
    async_load_b128(xs + lin4 * 4, src + base + tl * 512 + e4 * 4);
  }
  wait_async0();
  __syncthreads();

  v8f cre0 = {}, cim0 = {}, cre1 = {}, cim1 = {};
  for (int c = 0; c < 32; ++c) {
    int t0 = c * 32;
    int cur = c & 1;
    if (c + 1 < 32) {  // prefetch next chunk into the other buffer
#pragma unroll
      for (int i = 0; i < 2; ++i) {
        int lin4 = i * 256 + tid;
        int tl = lin4 >> 4, e4 = lin4 & 15;
        async_load_b128(xs + (cur ^ 1) * 2048 + lin4 * 4,
                        src + base + (t0 + 32 + tl) * 512 + e4 * 4);
      }
    }
    const float* xb = xs + cur * 2048;
#pragma unroll
    for (int ks = 0; ks < 8; ++ks) {
      int k0 = ks * 4;
      v2f a;
      a.x = xb[(k0 + 2 * hi) * 64 + et * 16 + lo];
      a.y = xb[(k0 + 2 * hi + 1) * 64 + et * 16 + lo];
      int tg = t0 + k0;
      v2f bc0 = load_b(FC, 64, tg, m0, lo, hi);
      v2f bs0 = load_b(FS, 64, tg, m0, lo, hi);
      v2f bc1 = load_b(FC, 64, tg, m1, lo, hi);
      v2f bs1 = load_b(FS, 64, tg, m1, lo, hi);
      cre0 = wmma4(a, bc0, cre0);
      cim0 = wmma4(a, bs0, cim0);
      cre1 = wmma4(a, bc1, cre1);
      cim1 = wmma4(a, bs1, cim1);
    }
    wait_async0();     // drain this wave's async writes to the other buffer
    __syncthreads();   // publish to all waves before next iteration uses it
  }

  if (!isk) {  // store transposed: [m][e]
    float* pr = ws + OFF_XQT_RE + bh * 4096;
    float* pi = ws + OFF_XQT_IM + bh * 4096;
#pragma unroll
    for (int v = 0; v < 8; ++v) {
      int e = et * 16 + v + 8 * hi;
      pr[(m0 + lo) * 64 + e] = cre0[v]; pi[(m0 + lo) * 64 + e] = cim0[v];
      pr[(m1 + lo) * 64 + e] = cre1[v]; pi[(m1 + lo) * 64 + e] = cim1[v];
    }
  } else {     // store [e][m]
    float* pr = ws + OFF_XK_RE + bh * 4096;
    float* pi = ws + OFF_XK_IM + bh * 4096;
#pragma unroll
    for (int v = 0; v < 8; ++v) {
      int e = et * 16 + v + 8 * hi;
      pr[e * 64 + m0 + lo] = cre0[v]; pi[e * 64 + m0 + lo] = cim0[v];
      pr[e * 64 + m1 + lo] = cre1[v]; pi[e * 64 + m1 + lo] = cim1[v];
    }
  }
}

// ================= K2: xqk = XqT * Xk (complex), then tanh =================
__global__ void k2_qk_tanh(float* __restrict__ ws) {
  int bh = blockIdx.x;
  const float* Ar = ws + OFF_XQT_RE + bh * 4096;
  const float* Ai = ws + OFF_XQT_IM + bh * 4096;
  const float* Br = ws + OFF_XK_RE + bh * 4096;
  const float* Bi = ws + OFF_XK_IM + bh * 4096;
  int lane = threadIdx.x & 31, w = threadIdx.x >> 5;
  int lo = lane & 15, hi = lane >> 4;
  int rt = w >> 1;                 // x tile
  int c0 = (w & 1) * 32, c1 = c0 + 16;  // y tiles

  v8f cre0 = {}, cim0 = {}, cre1 = {}, cim1 = {};
#pragma unroll
  for (int ks = 0; ks < 16; ++ks) {
    int k0 = ks * 4;
    v2f ar = load_a(Ar, 64, rt * 16, k0, lo, hi);
    v2f ai = load_a(Ai, 64, rt * 16, k0, lo, hi);
    v2f na = vneg(ai);
    v2f br0 = load_b(Br, 64, k0, c0, lo, hi);
    v2f bi0 = load_b(Bi, 64, k0, c0, lo, hi);
    cre0 = wmma4(ar, br0, cre0); cre0 = wmma4(na, bi0, cre0);
    cim0 = wmma4(ar, bi0, cim0); cim0 = wmma4(ai, br0, cim0);
    v2f br1 = load_b(Br, 64, k0, c1, lo, hi);
    v2f bi1 = load_b(Bi, 64, k0, c1, lo, hi);
    cre1 = wmma4(ar, br1, cre1); cre1 = wmma4(na, bi1, cre1);
    cim1 = wmma4(ar, bi1, cim1); cim1 = wmma4(ai, br1, cim1);
  }
  float* Tr = ws + OFF_T_RE + bh * 4096;
  float* Ti = ws + OFF_T_IM + bh * 4096;
#pragma unroll
  for (int v = 0; v < 8; ++v) {
    int x = rt * 16 + v + 8 * hi;
    float tr, ti;
    ctanh(cre0[v], cim0[v], tr, ti);
    Tr[x * 64 + c0 + lo] = tr; Ti[x * 64 + c0 + lo] = ti;
    ctanh(cre1[v], cim1[v], tr, ti);
    Tr[x * 64 + c1 + lo] = tr; Ti[x * 64 + c1 + lo] = ti;
  }
}

// ================= K3: xqkv[e,x] = sum_y Xk[e,y] * T[x,y] (complex) =========
__global__ void k3_attnv(float* __restrict__ ws) {
  int bh = blockIdx.x;
  int b = bh >> 3, h = bh & 7;
  const float* Ar = ws + OFF_XK_RE + bh * 4096;  // rows e, cols y
  const float* Ai = ws + OFF_XK_IM + bh * 4096;
  const float* Tr = ws + OFF_T_RE + bh * 4096;   // [x][y] -> use as B^T
  const float* Ti = ws + OFF_T_IM + bh * 4096;
  int lane = threadIdx.x & 31, w = threadIdx.x >> 5;
  int lo = lane & 15, hi = lane >> 4;
  int rt = w >> 1;                 // e tile
  int c0 = (w & 1) * 32, c1 = c0 + 16;  // x tiles

  v8f cre0 = {}, cim0 = {}, cre1 = {}, cim1 = {};
#pragma unroll
  for (int ks = 0; ks < 16; ++ks) {
    int k0 = ks * 4;
    v2f ar = load_a(Ar, 64, rt * 16, k0, lo, hi);
    v2f ai = load_a(Ai, 64, rt * 16, k0, lo, hi);
    v2f na = vneg(ai);
    v2f br0 = load_bT(Tr, 64, k0, c0, lo, hi);
    v2f bi0 = load_bT(Ti, 64, k0, c0, lo, hi);
    cre0 = wmma4(ar, br0, cre0); cre0 = wmma4(na, bi0, cre0);
    cim0 = wmma4(ar, bi0, cim0); cim0 = wmma4(ai, br0, cim0);
    v2f br1 = load_bT(Tr, 64, k0, c1, lo, hi);
    v2f bi1 = load_bT(Ti, 64, k0, c1, lo, hi);
    cre1 = wmma4(ar, br1, cre1); cre1 = wmma4(na, bi1, cre1);
    cim1 = wmma4(ar, bi1, cim1); cim1 = wmma4(ai, br1, cim1);
  }
  float* Vr = ws + OFF_V_RE;
  float* Vi = ws + OFF_V_IM;
#pragma unroll
  for (int v = 0; v < 8; ++v) {
    int e = rt * 16 + v + 8 * hi;
    int x0 = c0 + lo, x1 = c1 + lo;
    Vr[((h * 64 + x0) * 32 + b) * 64 + e] = cre0[v];
    Vi[((h * 64 + x0) * 32 + b) * 64 + e] = cim0[v];
    Vr[((h * 64 + x1) * 32 + b) * 64 + e] = cre1[v];
    Vi[((h * 64 + x1) * 32 + b) * 64 + e] = cim1[v];
  }
}

// ================= K4: Y[b,o] = sum_e V[b,e] * W[e,o]  per (h, mode x) ======
__global__ void k4_proj(const float* __restrict__ wf, float* __restrict__ ws) {
  int h = blockIdx.x >> 6, xm = blockIdx.x & 63;
  const float* Ar = ws + OFF_V_RE + (h * 64 + xm) * 2048;  // [b][e]
  const float* Ai = ws + OFF_V_IM + (h * 64 + xm) * 2048;
  int lane = threadIdx.x & 31, w = threadIdx.x >> 5;
  int lo = lane & 15, hi = lane >> 4;
  int rt = w >> 2;   // b tile 0..1
  int ct = w & 3;    // o tile 0..3

  v8f cre = {}, cim = {};
#pragma unroll
  for (int ks = 0; ks < 16; ++ks) {
    int k0 = ks * 4;
    v2f ar = load_a(Ar, 64, rt * 16, k0, lo, hi);
    v2f ai = load_a(Ai, 64, rt * 16, k0, lo, hi);
    v2f br, bi;
#pragma unroll
    for (int v = 0; v < 2; ++v) {
      int e = k0 + 2 * hi + v;
      int o = ct * 16 + lo;
      int cidx = ((h * 64 + e) * 64 + o) * 64 + xm;  // complex element index
      br[v] = wf[2 * cidx];
      bi[v] = wf[2 * cidx + 1];
    }
    cre = wmma4(ar, br, cre); cre = wmma4(vneg(ai), bi, cre);
    cim = wmma4(ar, bi, cim); cim = wmma4(ai, br, cim);
  }
  float* Y = ws + OFF_Y;
#pragma unroll
  for (int v = 0; v < 8; ++v) {
    int bb = rt * 16 + v + 8 * hi;
    int o = ct * 16 + lo;
    float* yb = Y + ((bb * 8 + h) * 64 + o) * 128;
    yb[xm] = cre[v];
    yb[64 + xm] = cim[v];
  }
}

// ================= K5: out = Y(64x128) * G(128x1024) per (b,h) ==============
__global__ void k5_idft(const float* __restrict__ ws, float* __restrict__ out) {
  __shared__ float yl[64 * 128];  // 32 KB
  int bh = blockIdx.x;
  const float* Yb = ws + OFF_Y + bh * 8192;
  // stage the Y slab with async b128 copies (512 float4s, 2 per thread)
#pragma unroll
  for (int i = 0; i < 2; ++i) {
    int lin4 = i * 256 + threadIdx.x;
    async_load_b128(yl + lin4 * 4, Yb + lin4 * 4);
  }
  wait_async0();
  __syncthreads();

  int lane = threadIdx.x & 31, w = threadIdx.x >> 5;
  int lo = lane & 15, hi = lane >> 4;
  int ot = w & 3;    // o tile
  int j0 = w >> 2;   // t-tile phase (0 or 1)
  const float* G = ws + OFF_G;

  for (int j = j0; j < 64; j += 2) {
    int t0 = j * 16;
    v8f acc = {};
#pragma unroll
    for (int ks = 0; ks < 32; ++ks) {
      int k0 = ks * 4;
      v2f a;
      a.x = yl[(ot * 16 + lo) * 128 + k0 + 2 * hi];
      a.y = yl[(ot * 16 + lo) * 128 + k0 + 2 * hi + 1];
      v2f b;
      b.x = G[(k0 + 2 * hi) * 1024 + t0 + lo];
      b.y = G[(k0 + 2 * hi + 1) * 1024 + t0 + lo];
      acc = wmma4(a, b, acc);
    }
#pragma unroll
    for (int v = 0; v < 8; ++v) {
      int o = ot * 16 + v + 8 * hi;
      out[bh * 65536 + o * 1024 + t0 + lo] = acc[v];
    }
  }
}

// ================= launch =================
extern "C" void kernel_launch(void* const* d_in, const int* in_sizes, int n_in,
                              void* d_out, int out_size, void* d_ws, size_t ws_size,
                              hipStream_t stream) {
  const float* q  = (const float*)d_in[0];
  const float* k  = (const float*)d_in[1];
  // d_in[2] = v (unused by the reference), d_in[4]/[5] = arange(64) indices
  const float* wf = (const float*)d_in[3];  // complex64 interleaved (re,im)
  float* ws = (float*)d_ws;
  float* out = (float*)d_out;

  k0_basis<<<512, 256, 0, stream>>>(ws);
  k1_dft<<<512, 256, 0, stream>>>(q, k, ws);
  k2_qk_tanh<<<256, 256, 0, stream>>>(ws);
  k3_attnv<<<256, 256, 0, stream>>>(ws);
  k4_proj<<<512, 256, 0, stream>>>(wf, ws);
  k5_idft<<<256, 256, 0, stream>>>(ws, out);
}